// Net_71287867179391
// MI455X (gfx1250) — compile-verified
//
#include <hip/hip_runtime.h>
#include <hip/hip_bf16.h>

typedef __attribute__((ext_vector_type(2))) float v2f;
typedef __attribute__((ext_vector_type(8))) float v8f;

#define GCN_B 8
#define GCN_N 50000
#define GCN_E 800000
#define GCN_F 32

// ---------------------------------------------------------------------------
// Generic zero-fill (grid-stride)
// ---------------------------------------------------------------------------
__global__ void zero_kernel(float* __restrict__ p, long long n) {
    long long i = (long long)blockIdx.x * blockDim.x + threadIdx.x;
    long long stride = (long long)gridDim.x * blockDim.x;
    for (; i < n; i += stride) p[i] = 0.0f;
}

// ---------------------------------------------------------------------------
// Degree accumulation over dst (self-loops folded in later as +1)
// ---------------------------------------------------------------------------
__global__ void degree_kernel(const int* __restrict__ ei, float* __restrict__ deg, int E) {
    int e = blockIdx.x * blockDim.x + threadIdx.x;
    if (e < E) atomicAdd(&deg[ei[E + e]], 1.0f);
}

// deg -> dinv = rsqrt(deg + 1)   (self-loop guarantees deg >= 1, no zero case)
__global__ void dinv_kernel(float* __restrict__ deg, int N) {
    int i = blockIdx.x * blockDim.x + threadIdx.x;
    if (i < N) deg[i] = rsqrtf(deg[i] + 1.0f);
}

// ---------------------------------------------------------------------------
// Dense transform: out[r, 0:32] = in[r, 0:32] @ W[32,32]  via V_WMMA_F32_16X16X4_F32
// One wave -> 16 rows x 32 cols (two 16x16 accumulators), K looped 8x4.
// A 16x4 fp32 layout: v0 = (lane<16 ? K=0 : K=2), v1 = (lane<16 ? K=1 : K=3), M=lane&15
// B 4x16 fp32 layout mirrored: N=lane&15, K base = (lane>=16)*2
// C/D 16x16: VGPR j holds M = j + 8*(lane>=16), N = lane&15
// ---------------------------------------------------------------------------
__global__ void gemm32_wmma(const float* __restrict__ in,
                            const float* __restrict__ W,
                            float* __restrict__ out, int nrows) {
    int wave = threadIdx.x >> 5;
    int lane = threadIdx.x & 31;
    int rowBase = (blockIdx.x * 8 + wave) * 16;
    if (rowBase >= nrows) return;                 // wave-uniform; EXEC stays all-ones
    int half = lane >> 4;                         // 0 or 1
    int l15  = lane & 15;
    const float* rowptr = in + (size_t)(rowBase + l15) * GCN_F;

    v8f acc0 = {};
    v8f acc1 = {};
#pragma unroll
    for (int kstep = 0; kstep < 8; ++kstep) {
        int k = kstep * 4 + half * 2;
        v2f a;  a.x  = rowptr[k];                 a.y  = rowptr[k + 1];
        v2f b0; b0.x = W[k * GCN_F + l15];        b0.y = W[(k + 1) * GCN_F + l15];
        v2f b1; b1.x = W[k * GCN_F + 16 + l15];   b1.y = W[(k + 1) * GCN_F + 16 + l15];
        acc0 = __builtin_amdgcn_wmma_f32_16x16x4_f32(false, a, false, b0,
                                                     (short)0, acc0, false, false);
        acc1 = __builtin_amdgcn_wmma_f32_16x16x4_f32(false, a, false, b1,
                                                     (short)0, acc1, false, false);
    }

    float* optr = out + (size_t)rowBase * GCN_F;
#pragma unroll
    for (int j = 0; j < 8; ++j) {
        int r = j + half * 8;
        optr[(size_t)r * GCN_F + l15]      = acc0[j];
        optr[(size_t)r * GCN_F + 16 + l15] = acc1[j];
    }
}

// ---------------------------------------------------------------------------
// Normalized scatter-add over edges (+self-loops), one thread per (edge, feat),
// unrolled over the 8 batches. f == lane so gathers/atomics are coalesced.
// ---------------------------------------------------------------------------
__global__ void agg_feat_kernel(const float* __restrict__ in,
                                float* __restrict__ out,
                                const int* __restrict__ ei,
                                const float* __restrict__ dinv,
                                int E, int N) {
    long long t = (long long)blockIdx.x * blockDim.x + threadIdx.x;
    long long total = (long long)(E + N) * GCN_F;
    if (t >= total) return;
    int f = (int)(t & (GCN_F - 1));
    int e = (int)(t >> 5);
    int s, d;
    if (e < E) { s = ei[e]; d = ei[E + e]; }
    else       { s = d = e - E; }
    float w = dinv[s] * dinv[d];
    size_t so   = (size_t)s * GCN_F + f;
    size_t dofs = (size_t)d * GCN_F + f;
    const size_t stride = (size_t)N * GCN_F;
#pragma unroll
    for (int b = 0; b < GCN_B; ++b)
        atomicAdd(out + dofs + (size_t)b * stride, in[so + (size_t)b * stride] * w);
}

// out[i] = relu(out[i] + bias[i % 32])
__global__ void bias_relu_kernel(float* __restrict__ buf,
                                 const float* __restrict__ bias, long long n) {
    long long i = (long long)blockIdx.x * blockDim.x + threadIdx.x;
    if (i < n) {
        float v = buf[i] + bias[i & (GCN_F - 1)];
        buf[i] = fmaxf(v, 0.0f);
    }
}

// s0[i] = dot(buf[i, 0:32], W3[0:32])
__global__ void dot32_kernel(const float* __restrict__ in,
                             const float* __restrict__ W3,
                             float* __restrict__ s0, long long nrows) {
    long long i = (long long)blockIdx.x * blockDim.x + threadIdx.x;
    if (i >= nrows) return;
    const float4* p = (const float4*)(in + i * GCN_F);
    const float4* w = (const float4*)W3;
    float acc = 0.0f;
#pragma unroll
    for (int k = 0; k < GCN_F / 4; ++k) {
        float4 a = p[k], b = w[k];
        acc += a.x * b.x + a.y * b.y + a.z * b.z + a.w * b.w;
    }
    s0[i] = acc;
}

// scalar scatter: s1[b, dst] += s0[b, src] * norm
__global__ void agg_scalar_kernel(const float* __restrict__ s0,
                                  float* __restrict__ s1,
                                  const int* __restrict__ ei,
                                  const float* __restrict__ dinv,
                                  int E, int N) {
    long long t = (long long)blockIdx.x * blockDim.x + threadIdx.x;
    long long total = (long long)(E + N) * GCN_B;
    if (t >= total) return;
    int b = (int)(t & (GCN_B - 1));
    int e = (int)(t >> 3);
    int s, d;
    if (e < E) { s = ei[e]; d = ei[E + e]; }
    else       { s = d = e - E; }
    float w = dinv[s] * dinv[d];
    atomicAdd(&s1[(size_t)b * N + d], s0[(size_t)b * N + s] * w);
}

// p = sigmoid(s1 + b3); out[b] = sum_n p   (block tree-reduce + atomic)
__global__ void sigmoid_reduce_kernel(const float* __restrict__ s1,
                                      const float* __restrict__ b3,
                                      float* __restrict__ out, int N) {
    __shared__ float sm[256];
    int b = blockIdx.y;
    int i = blockIdx.x * 256 + threadIdx.x;
    float v = 0.0f;
    if (i < N) {
        float z = s1[(size_t)b * N + i] + b3[0];
        v = 1.0f / (1.0f + expf(-z));
    }
    sm[threadIdx.x] = v;
    __syncthreads();
    for (int s = 128; s > 0; s >>= 1) {
        if (threadIdx.x < s) sm[threadIdx.x] += sm[threadIdx.x + s];
        __syncthreads();
    }
    if (threadIdx.x == 0) atomicAdd(&out[b], sm[0]);
}

// ---------------------------------------------------------------------------
extern "C" void kernel_launch(void* const* d_in, const int* in_sizes, int n_in,
                              void* d_out, int out_size, void* d_ws, size_t ws_size,
                              hipStream_t stream) {
    const float* x  = (const float*)d_in[0];
    const float* W1 = (const float*)d_in[1];
    const float* b1 = (const float*)d_in[2];
    const float* W2 = (const float*)d_in[3];
    const float* b2 = (const float*)d_in[4];
    const float* W3 = (const float*)d_in[5];
    const float* b3 = (const float*)d_in[6];
    const int*   ei = (const int*)d_in[7];
    float* out = (float*)d_out;

    const int B = GCN_B, N = GCN_N, E = GCN_E, F = GCN_F;
    const long long BN  = (long long)B * N;         // 400k rows
    const long long BNF = BN * F;                   // 12.8M floats

    // workspace layout (floats)
    float* ws   = (float*)d_ws;
    float* dinv = ws;                 // N   (deg in-place -> dinv)
    float* buf0 = dinv + N;           // BNF
    float* buf1 = buf0 + BNF;         // BNF
    float* s0   = buf1 + BNF;         // BN
    float* s1   = s0 + BN;            // BN

    const int T = 256;
    const long long featWork   = (long long)(E + N) * F;
    const long long scalarWork = (long long)(E + N) * B;
    dim3 gGemm((unsigned)((BN / 16 + 7) / 8));              // 8 waves/block, 16 rows/wave
    dim3 gFeat((unsigned)((featWork + T - 1) / T));
    dim3 gScal((unsigned)((scalarWork + T - 1) / T));
    dim3 gElem((unsigned)((BNF + T - 1) / T));
    dim3 gRows((unsigned)((BN + T - 1) / T));
    dim3 gRed((unsigned)((N + T - 1) / T), B);

    // normalization
    zero_kernel<<<64, T, 0, stream>>>(dinv, N);
    zero_kernel<<<1, T, 0, stream>>>(out, B);
    degree_kernel<<<(E + T - 1) / T, T, 0, stream>>>(ei, dinv, E);
    dinv_kernel<<<(N + T - 1) / T, T, 0, stream>>>(dinv, N);

    // layer 1
    gemm32_wmma<<<gGemm, T, 0, stream>>>(x, W1, buf0, (int)BN);
    zero_kernel<<<2048, T, 0, stream>>>(buf1, BNF);
    agg_feat_kernel<<<gFeat, T, 0, stream>>>(buf0, buf1, ei, dinv, E, N);
    bias_relu_kernel<<<gElem, T, 0, stream>>>(buf1, b1, BNF);

    // layer 2
    gemm32_wmma<<<gGemm, T, 0, stream>>>(buf1, W2, buf0, (int)BN);
    zero_kernel<<<2048, T, 0, stream>>>(buf1, BNF);
    agg_feat_kernel<<<gFeat, T, 0, stream>>>(buf0, buf1, ei, dinv, E, N);
    bias_relu_kernel<<<gElem, T, 0, stream>>>(buf1, b2, BNF);

    // layer 3 (projection to scalar) + aggregation + sigmoid-sum
    dot32_kernel<<<gRows, T, 0, stream>>>(buf1, W3, s0, BN);
    zero_kernel<<<512, T, 0, stream>>>(s1, BN);
    agg_scalar_kernel<<<gScal, T, 0, stream>>>(s0, s1, ei, dinv, E, N);
    sigmoid_reduce_kernel<<<gRed, T, 0, stream>>>(s1, b3, out, N);
}